// GResBlock_26800595927569
// MI455X (gfx1250) — compile-verified
//
#include <hip/hip_runtime.h>

typedef __attribute__((ext_vector_type(2))) float v2f;
typedef __attribute__((ext_vector_type(8))) float v8f;

#define N_NODES 50000
#define E_EDGES 1600000
#define F 128  // IN == HID == 128

// ---------------- degree / normalization ----------------

__global__ void k_init_deg(float* __restrict__ deg) {
    int i = blockIdx.x * blockDim.x + threadIdx.x;
    if (i < N_NODES) deg[i] = 1.0f;  // self-loop contribution
}

__global__ void k_deg(const long long* __restrict__ dst, float* __restrict__ deg) {
    int e = blockIdx.x * blockDim.x + threadIdx.x;
    if (e < E_EDGES) unsafeAtomicAdd(&deg[(int)dst[e]], 1.0f);
}

__global__ void k_rsqrt_inplace(float* __restrict__ deg) {
    int i = blockIdx.x * blockDim.x + threadIdx.x;
    if (i < N_NODES) deg[i] = rsqrtf(fmaxf(deg[i], 1.0f));
}

// ---------------- zero fill ----------------

__global__ void k_zero4(float4* __restrict__ p, long n4) {
    long i = (long)blockIdx.x * blockDim.x + threadIdx.x;
    if (i < n4) p[i] = make_float4(0.f, 0.f, 0.f, 0.f);
}

// ---------------- dense GEMM: H[N,128] = X[N,128] @ W[128,128] ----------------
// One block = one 16-row tile of X; 8 waves cover the 8 column tiles of W.
// W staged in LDS (64 KB). fp32 WMMA 16x16x4, 32 k-steps.

__global__ __launch_bounds__(256) void k_gemm128(const float* __restrict__ X,
                                                 const float* __restrict__ W,
                                                 float* __restrict__ H) {
    __shared__ float Wlds[F * F];
    int tid = threadIdx.x;

    // cooperative load of W (16384 floats) as float4
    const float4* Wv = (const float4*)W;
    float4* Lv = (float4*)Wlds;
#pragma unroll
    for (int i = 0; i < 16; ++i) Lv[tid + i * 256] = Wv[tid + i * 256];
    __syncthreads();

    int wave = tid >> 5;   // 0..7 -> column tile
    int lane = tid & 31;
    int m0 = blockIdx.x * 16;       // row tile base (N=50000 is 3125*16 exactly)
    int n0 = wave * 16;             // col tile base
    int mrow = m0 + (lane & 15);    // A: M per lane
    int khalf = (lane >> 4) * 2;    // A/B: lanes 16-31 hold K+2,K+3
    int ncol = n0 + (lane & 15);    // B: N per lane

    const float* xrow = X + (long)mrow * F;
    v8f acc = {};
#pragma unroll 8
    for (int k0 = 0; k0 < F; k0 += 4) {
        v2f a, b;
        a.x = xrow[k0 + khalf];
        a.y = xrow[k0 + khalf + 1];
        b.x = Wlds[(k0 + khalf) * F + ncol];
        b.y = Wlds[(k0 + khalf + 1) * F + ncol];
        acc = __builtin_amdgcn_wmma_f32_16x16x4_f32(
            /*neg_a=*/false, a, /*neg_b=*/false, b,
            /*c_mod=*/(short)0, acc, /*reuse_a=*/false, /*reuse_b=*/false);
    }

    // C/D layout: VGPR r -> M = r (lanes 0-15) / r+8 (lanes 16-31), N = lane&15
    int mbase = m0 + (lane >> 4) * 8;
    int nc = n0 + (lane & 15);
#pragma unroll
    for (int r = 0; r < 8; ++r) H[(long)(mbase + r) * F + nc] = acc[r];
}

// ---------------- edge scatter: Out[dst] += Hin[src] * dis[src]*dis[dst] ----------------
// One wave32 per edge; each lane handles a float4 (32*4 = 128 features).

__global__ __launch_bounds__(256) void k_agg(const long long* __restrict__ src,
                                             const long long* __restrict__ dst,
                                             const float* __restrict__ dis,
                                             const float* __restrict__ Hin,
                                             float* __restrict__ Out) {
    int e = blockIdx.x * (blockDim.x >> 5) + (threadIdx.x >> 5);
    if (e >= E_EDGES) return;
    int lane = threadIdx.x & 31;
    int s = (int)src[e];
    int d = (int)dst[e];
    float w = dis[s] * dis[d];
    float4 v = ((const float4*)(Hin + (long)s * F))[lane];
    float* op = Out + (long)d * F + lane * 4;
    unsafeAtomicAdd(op + 0, v.x * w);
    unsafeAtomicAdd(op + 1, v.y * w);
    unsafeAtomicAdd(op + 2, v.z * w);
    unsafeAtomicAdd(op + 3, v.w * w);
}

// ---------------- epilogues (self-loop + bias + relu [+ residual]) ----------------

__global__ void k_post_relu(const float* __restrict__ agg, const float* __restrict__ h,
                            const float* __restrict__ b, const float* __restrict__ dis,
                            float* __restrict__ outH) {
    long i = (long)blockIdx.x * blockDim.x + threadIdx.x;
    if (i >= (long)N_NODES * F) return;
    int node = (int)(i >> 7);
    int f = (int)(i & 127);
    float sl = dis[node];
    sl *= sl;
    float v = agg[i] + sl * h[i] + b[f];
    outH[i] = fmaxf(v, 0.0f);
}

__global__ void k_post_final(const float* __restrict__ x, const float* __restrict__ h2,
                             const float* __restrict__ b, const float* __restrict__ dis,
                             float* __restrict__ out) {  // out holds agg2 on entry
    long i = (long)blockIdx.x * blockDim.x + threadIdx.x;
    if (i >= (long)N_NODES * F) return;
    int node = (int)(i >> 7);
    int f = (int)(i & 127);
    float sl = dis[node];
    sl *= sl;
    float v = out[i] + sl * h2[i] + b[f];
    out[i] = (x[i] + fmaxf(v, 0.0f)) * 0.5f;
}

// ---------------- launch ----------------

extern "C" void kernel_launch(void* const* d_in, const int* in_sizes, int n_in,
                              void* d_out, int out_size, void* d_ws, size_t ws_size,
                              hipStream_t stream) {
    const float* x = (const float*)d_in[0];
    const long long* edges = (const long long*)d_in[1];  // int64, shape (2, E)
    const float* W1 = (const float*)d_in[2];
    const float* b1 = (const float*)d_in[3];
    const float* W2 = (const float*)d_in[4];
    const float* b2 = (const float*)d_in[5];
    float* out = (float*)d_out;

    const long long* esrc = edges;
    const long long* edst = edges + E_EDGES;

    float* dis = (float*)d_ws;                 // [N]
    float* A = dis + N_NODES;                  // [N*F]
    float* B = A + (long)N_NODES * F;          // [N*F]

    const long NF = (long)N_NODES * F;
    dim3 blk(256);
    int gN = (N_NODES + 255) / 256;
    int gE = (E_EDGES + 255) / 256;
    int gNF = (int)((NF + 255) / 256);
    int gZ = (int)((NF / 4 + 255) / 256);
    int gGemm = N_NODES / 16;          // 3125 row tiles
    int gAgg = (E_EDGES + 7) / 8;      // 8 edges (waves) per block

    // normalization: deg -> dis = rsqrt(deg)
    k_init_deg<<<gN, blk, 0, stream>>>(dis);
    k_deg<<<gE, blk, 0, stream>>>(edst, dis);
    k_rsqrt_inplace<<<gN, blk, 0, stream>>>(dis);

    // layer 1
    k_gemm128<<<gGemm, blk, 0, stream>>>(x, W1, A);
    k_zero4<<<gZ, blk, 0, stream>>>((float4*)B, NF / 4);
    k_agg<<<gAgg, blk, 0, stream>>>(esrc, edst, dis, A, B);
    k_post_relu<<<gNF, blk, 0, stream>>>(B, A, b1, dis, A);  // A = relu(...)

    // layer 2
    k_gemm128<<<gGemm, blk, 0, stream>>>(A, W2, B);
    k_zero4<<<gZ, blk, 0, stream>>>((float4*)out, NF / 4);
    k_agg<<<gAgg, blk, 0, stream>>>(esrc, edst, dis, B, out);
    k_post_final<<<gNF, blk, 0, stream>>>(x, B, b2, dis, out);
}